// GaussianRasterizer_50672024158377
// MI455X (gfx1250) — compile-verified
//
#include <hip/hip_runtime.h>
#include <hip/hip_bf16.h>
#include <math.h>

typedef __attribute__((ext_vector_type(2))) float v2f;
typedef __attribute__((ext_vector_type(4))) float v4f;
typedef __attribute__((ext_vector_type(8))) float v8f;

#define NGAUSS   2048
#define NVIEW    2
#define IMG_H    80
#define IMG_W    144
#define HWPIX    (IMG_H * IMG_W)          // 11520
#define NCHAN    32
#define TILES_PER_VIEW (HWPIX / 16)       // 720
#define NEAR_P     0.2f
#define DILATE_P   0.3f
#define ALPHA_MIN_P (1.0f / 255.0f)
#define T_EPS_P    1e-4f

// ---------------------------------------------------------------------------
// Workspace layout (floats):
//   raw  : 8 SoA arrays [px,py,ca,cb,cc,tz,op,key] x [NVIEW][NGAUSS]
//   spA  : sorted AoS records, 8 floats per gaussian: {px,py,ca,cb, cc,tz,op,0}
//          -> two b128 loads per gaussian in the hot loop
//   sidx : int [NVIEW][NGAUSS]  (sorted -> original index, for color fetch)
//   nvalid: int [NVIEW]
// ---------------------------------------------------------------------------

// ===========================================================================
// Kernel 1: per-(view,gaussian) projection + 2D covariance.
// ===========================================================================
__global__ void gs_preprocess(const float* __restrict__ means3d,
                              const float* __restrict__ opacities,
                              const float* __restrict__ scales,
                              const float* __restrict__ rotations,
                              const float* __restrict__ cam2img,
                              const float* __restrict__ cam2ego,
                              float* __restrict__ raw) {
  int id = blockIdx.x * blockDim.x + threadIdx.x;
  if (id >= NVIEW * NGAUSS) return;
  int v = id / NGAUSS;
  int g = id % NGAUSS;

  const float* K   = cam2img + v * 9;
  const float* c2e = cam2ego + v * 16;
  float fx = K[0], cx = K[2], fy = K[4], cy = K[5];

  // R = c2e[:3,:3]^T ; Tt = -R @ c2e[:3,3]
  float Rm[3][3];
#pragma unroll
  for (int i = 0; i < 3; ++i)
#pragma unroll
    for (int j = 0; j < 3; ++j) Rm[i][j] = c2e[j * 4 + i];
  float Tt[3];
#pragma unroll
  for (int i = 0; i < 3; ++i)
    Tt[i] = -(Rm[i][0] * c2e[3] + Rm[i][1] * c2e[7] + Rm[i][2] * c2e[11]);

  float m0 = means3d[g * 3 + 0], m1 = means3d[g * 3 + 1], m2 = means3d[g * 3 + 2];
  float t0 = Rm[0][0] * m0 + Rm[0][1] * m1 + Rm[0][2] * m2 + Tt[0];
  float t1 = Rm[1][0] * m0 + Rm[1][1] * m1 + Rm[1][2] * m2 + Tt[1];
  float t2 = Rm[2][0] * m0 + Rm[2][1] * m1 + Rm[2][2] * m2 + Tt[2];

  // quaternion -> rotation
  float qw = rotations[g * 4 + 0], qx = rotations[g * 4 + 1];
  float qy = rotations[g * 4 + 2], qz = rotations[g * 4 + 3];
  float qn = rsqrtf(qw * qw + qx * qx + qy * qy + qz * qz);
  qw *= qn; qx *= qn; qy *= qn; qz *= qn;
  float Rq[3][3];
  Rq[0][0] = 1.f - 2.f * (qy * qy + qz * qz);
  Rq[0][1] = 2.f * (qx * qy - qw * qz);
  Rq[0][2] = 2.f * (qx * qz + qw * qy);
  Rq[1][0] = 2.f * (qx * qy + qw * qz);
  Rq[1][1] = 1.f - 2.f * (qx * qx + qz * qz);
  Rq[1][2] = 2.f * (qy * qz - qw * qx);
  Rq[2][0] = 2.f * (qx * qz - qw * qy);
  Rq[2][1] = 2.f * (qy * qz + qw * qx);
  Rq[2][2] = 1.f - 2.f * (qx * qx + qy * qy);

  float s0 = scales[g * 3 + 0], s1 = scales[g * 3 + 1], s2 = scales[g * 3 + 2];
  float M[3][3];
#pragma unroll
  for (int i = 0; i < 3; ++i) {
    M[i][0] = Rq[i][0] * s0; M[i][1] = Rq[i][1] * s1; M[i][2] = Rq[i][2] * s2;
  }
  float Sig[3][3];
#pragma unroll
  for (int i = 0; i < 3; ++i)
#pragma unroll
    for (int j = 0; j < 3; ++j)
      Sig[i][j] = M[i][0] * M[j][0] + M[i][1] * M[j][1] + M[i][2] * M[j][2];

  // covcam = Rm * Sig * Rm^T
  float tmp[3][3], cvc[3][3];
#pragma unroll
  for (int i = 0; i < 3; ++i)
#pragma unroll
    for (int b = 0; b < 3; ++b)
      tmp[i][b] = Rm[i][0] * Sig[0][b] + Rm[i][1] * Sig[1][b] + Rm[i][2] * Sig[2][b];
#pragma unroll
  for (int i = 0; i < 3; ++i)
#pragma unroll
    for (int j = 0; j < 3; ++j)
      cvc[i][j] = tmp[i][0] * Rm[j][0] + tmp[i][1] * Rm[j][1] + tmp[i][2] * Rm[j][2];

  bool valid = t2 > NEAR_P;
  float tzs = valid ? t2 : 1.0f;
  float tanfovx = (float)IMG_W / (2.0f * fx);
  float tanfovy = (float)IMG_H / (2.0f * fy);
  float txtz = fminf(fmaxf(t0 / tzs, -1.3f * tanfovx), 1.3f * tanfovx);
  float tytz = fminf(fmaxf(t1 / tzs, -1.3f * tanfovy), 1.3f * tanfovy);

  float J00 = fx / tzs, J02 = -fx * txtz / tzs;
  float J11 = fy / tzs, J12 = -fy * tytz / tzs;

  // cov2 = J covcam J^T  (2x2, symmetric)
  float u0 = J00 * cvc[0][0] + J02 * cvc[2][0];
  float u1 = J00 * cvc[0][1] + J02 * cvc[2][1];
  float u2 = J00 * cvc[0][2] + J02 * cvc[2][2];
  float w1 = J11 * cvc[1][1] + J12 * cvc[2][1];
  float w2 = J11 * cvc[1][2] + J12 * cvc[2][2];
  float a = u0 * J00 + u2 * J02 + DILATE_P;
  float b = u1 * J11 + u2 * J12;
  float c = w1 * J11 + w2 * J12 + DILATE_P;
  float det = a * c - b * b;
  det = (det == 0.0f) ? 1.0f : det;
  float inv = 1.0f / det;
  float ca = c * inv, cb = -b * inv, cc = a * inv;

  float px = fx * t0 / tzs + ((float)IMG_W - cx) - 0.5f;
  float py = fy * t1 / tzs + ((float)IMG_H - cy) - 0.5f;
  float key = valid ? t2 : __builtin_inff();

  raw[(0 * NVIEW + v) * NGAUSS + g] = px;
  raw[(1 * NVIEW + v) * NGAUSS + g] = py;
  raw[(2 * NVIEW + v) * NGAUSS + g] = ca;
  raw[(3 * NVIEW + v) * NGAUSS + g] = cb;
  raw[(4 * NVIEW + v) * NGAUSS + g] = cc;
  raw[(5 * NVIEW + v) * NGAUSS + g] = t2;
  raw[(6 * NVIEW + v) * NGAUSS + g] = opacities[g];
  raw[(7 * NVIEW + v) * NGAUSS + g] = key;
}

// ===========================================================================
// Kernel 2: per-view in-LDS bitonic sort (2048 keys) + AoS gather.
// One workgroup of 1024 threads per view.  16KB LDS of the 320KB/WGP pool.
// ===========================================================================
__global__ __launch_bounds__(1024) void gs_sort(const float* __restrict__ raw,
                                                float* __restrict__ spA,
                                                int* __restrict__ sidx,
                                                int* __restrict__ nvalid) {
  __shared__ float k_[NGAUSS];
  __shared__ int   id_[NGAUSS];
  int v   = blockIdx.x;
  int tid = threadIdx.x;
  const float* raw_key = raw + (7 * NVIEW + v) * NGAUSS;

  for (int t = tid; t < NGAUSS; t += 1024) { k_[t] = raw_key[t]; id_[t] = t; }

  for (int k = 2; k <= NGAUSS; k <<= 1) {
    for (int j = k >> 1; j > 0; j >>= 1) {
      __syncthreads();
      for (int t = tid; t < NGAUSS; t += 1024) {
        int ixj = t ^ j;
        if (ixj > t) {
          bool up = ((t & k) == 0);
          float ka = k_[t], kb = k_[ixj];
          bool sw = up ? (ka > kb) : (ka < kb);
          if (sw) {
            k_[t] = kb; k_[ixj] = ka;
            int ti = id_[t]; id_[t] = id_[ixj]; id_[ixj] = ti;
          }
        }
      }
    }
  }
  __syncthreads();

  // nvalid = index of first +inf key (invalids sorted to the end)
  for (int t = tid; t < NGAUSS; t += 1024) {
    bool curInf = isinf(k_[t]);
    if (curInf && (t == 0 || !isinf(k_[t - 1]))) nvalid[v] = t;
  }
  if (tid == 0 && !isinf(k_[NGAUSS - 1])) nvalid[v] = NGAUSS;

  // gather sorted params into AoS records of 8 floats
  for (int t = tid; t < NGAUSS; t += 1024) {
    int s = id_[t];
    sidx[v * NGAUSS + t] = s;
    float* rec = spA + (v * NGAUSS + t) * 8;
#pragma unroll
    for (int arr = 0; arr < 7; ++arr)
      rec[arr] = raw[(arr * NVIEW + v) * NGAUSS + s];
    rec[7] = 0.0f;
  }
}

// ===========================================================================
// Kernel 3: rasterizer.  One wave = 16 pixels (WMMA M-tile).
// Gaussian evaluation (power + v_exp_f32, the VALU bottleneck) is split
// across the half-waves: lanes 0-15 evaluate chunk gaussians {k0,k0+1},
// lanes 16-31 evaluate {k0+2,k0+3} for the SAME pixel, then the pair
// exchanges alphas/tz via __shfl_xor(16) and both replay the cheap
// sequential transmittance chain.  Color accumulation goes through
// V_WMMA_F32_16X16X4_F32 (two N-tiles for 32 channels) in native f32.
// ===========================================================================
__global__ __launch_bounds__(256) void gs_raster(const float* __restrict__ colors,
                                                 const float* __restrict__ spA,
                                                 const int* __restrict__ sidx,
                                                 const int* __restrict__ nvalid,
                                                 float* __restrict__ out) {
  int lane   = threadIdx.x & 31;
  int waveId = threadIdx.x >> 5;
  int tile   = blockIdx.x * 8 + waveId;
  if (tile >= NVIEW * TILES_PER_VIEW) return;
  int view      = tile / TILES_PER_VIEW;
  int tile_base = (tile % TILES_PER_VIEW) * 16;

  bool hi   = lane >= 16;
  int  chan = lane & 15;           // channel column for B/D; also pixel row
  int  pix  = tile_base + chan;    // pixel owned by this lane pair
  float X = (float)(pix % IMG_W);
  float Y = (float)(pix / IMG_W);

  const v4f* sp4 = (const v4f*)(spA + (size_t)view * NGAUSS * 8);
  const int* sV  = sidx + view * NGAUSS;
  int nv = nvalid[view];

  v8f acc0 = {};           // channels 0..15
  v8f acc1 = {};           // channels 16..31
  float dep = 0.0f;
  float T = 1.0f;
  bool done = false;

  for (int k0 = 0; k0 < nv; k0 += 4) {
    // this lane evaluates 2 gaussians: lo half -> k0,k0+1 ; hi half -> k0+2,k0+3
    int gbase = k0 + (hi ? 2 : 0);
    v4f q0 = sp4[gbase * 2 + 0];   // px,py,ca,cb   (gaussian gbase)
    v4f q1 = sp4[gbase * 2 + 1];   // cc,tz,op,0
    v4f r0 = sp4[gbase * 2 + 2];   // px,py,ca,cb   (gaussian gbase+1)
    v4f r1 = sp4[gbase * 2 + 3];   // cc,tz,op,0

    float dx0 = X - q0.x, dy0 = Y - q0.y;
    float pw0 = -0.5f * (q0.z * dx0 * dx0 + q1.x * dy0 * dy0) - q0.w * dx0 * dy0;
    float e0  = fminf(0.99f, q1.z * __expf(pw0));
    e0 = ((pw0 > 0.0f) || (e0 < ALPHA_MIN_P) || (gbase >= nv)) ? 0.0f : e0;

    float dx1 = X - r0.x, dy1 = Y - r0.y;
    float pw1 = -0.5f * (r0.z * dx1 * dx1 + r1.x * dy1 * dy1) - r0.w * dx1 * dy1;
    float e1  = fminf(0.99f, r1.z * __expf(pw1));
    e1 = ((pw1 > 0.0f) || (e1 < ALPHA_MIN_P) || (gbase + 1 >= nv)) ? 0.0f : e1;

    // exchange alphas + depths with the partner lane (same pixel, other ks)
    float xe0 = __shfl_xor(e0, 16, 32);
    float xe1 = __shfl_xor(e1, 16, 32);
    float xt0 = __shfl_xor(q1.y, 16, 32);
    float xt1 = __shfl_xor(r1.y, 16, 32);

    float al[4], tzk[4];
    al[0]  = hi ? xe0  : e0;   al[1]  = hi ? xe1  : e1;
    al[2]  = hi ? e0   : xe0;  al[3]  = hi ? e1   : xe1;
    tzk[0] = hi ? xt0  : q1.y; tzk[1] = hi ? xt1  : r1.y;
    tzk[2] = hi ? q1.y : xt0;  tzk[3] = hi ? r1.y : xt1;

    // sequential transmittance chain (cheap; replayed identically per pair)
    float wgt[4];
#pragma unroll
    for (int k = 0; k < 4; ++k) {
      float alpha = al[k];
      float Tn = T * (1.0f - alpha);
      float wk = (Tn >= T_EPS_P) ? alpha * T : 0.0f;
      done = done || (Tn < T_EPS_P);
      T = Tn;
      dep += wk * tzk[k];
      wgt[k] = wk;
    }

    // A (16x4 f32): VGPR0 = K0 (lanes 0-15) / K2 (lanes 16-31); VGPR1 = K1/K3
    v2f a;
    a.x = hi ? wgt[2] : wgt[0];
    a.y = hi ? wgt[3] : wgt[1];

    // B (4x16 f32): VGPR0 = row K0 / row K2 ; VGPR1 = row K1 / row K3
    int s0 = sV[gbase], s1 = sV[gbase + 1];
    v2f b0, b1;
    b0.x = colors[s0 * NCHAN + chan];
    b0.y = colors[s1 * NCHAN + chan];
    b1.x = colors[s0 * NCHAN + chan + 16];
    b1.y = colors[s1 * NCHAN + chan + 16];

    acc0 = __builtin_amdgcn_wmma_f32_16x16x4_f32(false, a, false, b0,
                                                 (short)0, acc0, false, false);
    acc1 = __builtin_amdgcn_wmma_f32_16x16x4_f32(false, a, false, b1,
                                                 (short)0, acc1, false, false);

    if (__all((int)done)) break;   // transmittance saturated for all 16 pixels
  }

  // D layout: VGPR r -> M=r (lanes 0-15) / M=8+r (lanes 16-31), N=lane&15
  float* rendered = out;                           // (V, 32, H, W)
  float* depth    = out + NVIEW * NCHAN * HWPIX;   // (V, 1, H, W)
#pragma unroll
  for (int r = 0; r < 8; ++r) {
    int mrow = hi ? (8 + r) : r;
    int p = tile_base + mrow;
    rendered[(view * NCHAN + chan) * HWPIX + p]        = acc0[r];
    rendered[(view * NCHAN + chan + 16) * HWPIX + p]   = acc1[r];
  }
  if (lane < 16) depth[view * HWPIX + tile_base + lane] = dep;
}

// ===========================================================================
extern "C" void kernel_launch(void* const* d_in, const int* in_sizes, int n_in,
                              void* d_out, int out_size, void* d_ws, size_t ws_size,
                              hipStream_t stream) {
  const float* means3d   = (const float*)d_in[0];
  const float* colors    = (const float*)d_in[1];
  const float* opacities = (const float*)d_in[2];
  const float* scales    = (const float*)d_in[3];
  const float* rotations = (const float*)d_in[4];
  const float* cam2img   = (const float*)d_in[5];
  const float* cam2ego   = (const float*)d_in[6];
  (void)in_sizes; (void)n_in; (void)out_size; (void)ws_size;

  float* raw    = (float*)d_ws;                       // 8 * V * N floats (SoA)
  float* spA    = raw + 8 * NVIEW * NGAUSS;           // 8 * V * N floats (AoS)
  int*   sidx   = (int*)(spA + 8 * NVIEW * NGAUSS);   // V * N ints
  int*   nvalid = sidx + NVIEW * NGAUSS;              // V ints

  float* out = (float*)d_out;

  int pre_threads = NVIEW * NGAUSS;
  gs_preprocess<<<(pre_threads + 255) / 256, 256, 0, stream>>>(
      means3d, opacities, scales, rotations, cam2img, cam2ego, raw);

  gs_sort<<<NVIEW, 1024, 0, stream>>>(raw, spA, sidx, nvalid);

  int total_tiles = NVIEW * TILES_PER_VIEW;           // 1440 waves, 8 per block
  gs_raster<<<(total_tiles + 7) / 8, 256, 0, stream>>>(colors, spA, sidx, nvalid, out);
}